// SoftDTW_64089501990976
// MI455X (gfx1250) — compile-verified
//
#include <hip/hip_runtime.h>
#include <math.h>

typedef __attribute__((ext_vector_type(2))) float v2f;
typedef __attribute__((ext_vector_type(8))) float v8f;

#define BATCH 32
#define NPTS  1024
#define MPTS  1024
#define DIMK  32
#define BIGV  1e10f

// ---------------------------------------------------------------------------
// Kernel 1: row squared-norms for X and Y.  idx < B*N handles X, rest Y.
// ---------------------------------------------------------------------------
__global__ void __launch_bounds__(256)
rownorm_kernel(const float* __restrict__ X, const float* __restrict__ Y,
               float* __restrict__ x2, float* __restrict__ y2) {
    int idx = blockIdx.x * blockDim.x + threadIdx.x;      // 0 .. 2*B*N-1
    const int total = BATCH * NPTS;
    const float* src;
    float* dst;
    int r;
    if (idx < total) { src = X; dst = x2; r = idx; }
    else             { src = Y; dst = y2; r = idx - total; }
    const float4* row = (const float4*)(src + (size_t)r * DIMK);
    float s = 0.0f;
#pragma unroll
    for (int q = 0; q < DIMK / 4; ++q) {
        float4 v = row[q];
        s += v.x * v.x + v.y * v.y + v.z * v.z + v.w * v.w;
    }
    dst[r] = s;
}

// ---------------------------------------------------------------------------
// Kernel 2: D[b,n,m] = x2[n] + y2[m] - 2 * (X Y^T) via V_WMMA_F32_16X16X4_F32.
// One wave per 16x16 tile; K=32 -> 8 chained WMMAs.
//
// f32 16x16x4 lane layout (ISA 7.12.2):
//   A (16x4):  lane L -> row M = L%16, a[v] = A[M][2*(L/16)+v]
//   B (4x16):  lane L -> col N = L%16, b[v] = B[2*(L/16)+v][N]
//   C (16x16): lane L -> acc[r] = C[8*(L/16)+r][L%16]
// Both A and B loads are aligned float2 from k-contiguous rows of X / Y.
// ---------------------------------------------------------------------------
__global__ void __launch_bounds__(256)
dist_wmma_kernel(const float* __restrict__ X, const float* __restrict__ Y,
                 const float* __restrict__ x2, const float* __restrict__ y2,
                 float* __restrict__ Dm) {
    const int wavesPerBlock = blockDim.x >> 5;
    int wave = blockIdx.x * wavesPerBlock + (threadIdx.x >> 5);
    int lane = threadIdx.x & 31;

    // 64x64 = 4096 tiles per batch
    int b  = wave >> 12;
    int t  = wave & 4095;
    int n0 = (t >> 6) << 4;
    int m0 = (t & 63) << 4;

    int lo = lane & 15;      // row/col within tile
    int hi = lane >> 4;      // K sub-group / M-half selector

    const float* xrow = X + ((size_t)b * NPTS + n0 + lo) * DIMK + hi * 2;
    const float* yrow = Y + ((size_t)b * MPTS + m0 + lo) * DIMK + hi * 2;

    v8f acc = {};
#pragma unroll
    for (int kc = 0; kc < DIMK / 4; ++kc) {
        v2f a  = *(const v2f*)(xrow + kc * 4);
        v2f bb = *(const v2f*)(yrow + kc * 4);
        acc = __builtin_amdgcn_wmma_f32_16x16x4_f32(
            /*neg_a=*/false, a, /*neg_b=*/false, bb,
            /*c_mod=*/(short)0, acc, /*reuse_a=*/false, /*reuse_b=*/false);
    }

    float yv = y2[b * MPTS + m0 + lo];
    const float* xs = x2 + b * NPTS + n0 + hi * 8;
    float* dst = Dm + ((size_t)b * NPTS + n0 + hi * 8) * MPTS + m0 + lo;
#pragma unroll
    for (int r = 0; r < 8; ++r) {
        dst[(size_t)r * MPTS] = xs[r] + yv - 2.0f * acc[r];
    }
}

// ---------------------------------------------------------------------------
// Kernel 3: soft-DTW anti-diagonal wavefront.  One workgroup (1024 threads =
// 32 wave32s) per batch.  Three DP rows rotate in LDS; thread i (=tid+1)
// owns DP column i and streams row i-1 of D with unit stride as the diagonal
// advances (L2-resident: D is 128 MB < 192 MB L2).
// ---------------------------------------------------------------------------
__global__ void __launch_bounds__(1024)
sdtw_kernel(const float* __restrict__ Dm, float* __restrict__ out) {
    __shared__ float R[3][NPTS + 1];
    const int b   = blockIdx.x;
    const int tid = threadIdx.x;
    const int i   = tid + 1;                     // DP column 1..N

    R[0][tid] = BIGV; R[1][tid] = BIGV; R[2][tid] = BIGV;
    if (tid == 0) {
        R[0][NPTS] = BIGV; R[1][NPTS] = BIGV; R[2][NPTS] = BIGV;
        R[0][0] = 0.0f;                          // rm2[0] = 0
    }
    __syncthreads();

    int r2 = 0, r1 = 1, rn = 2;
    const float* drow = Dm + ((size_t)b * NPTS + (i - 1)) * MPTS;
    float result = 0.0f;

    for (int d = 2; d <= NPTS + MPTS; ++d) {
        int j = d - i;
        float nv = BIGV;
        if (j >= 1 && j <= MPTS) {
            float a = R[r2][i - 1];              // NW
            float g = R[r1][i - 1];              // N
            float c = R[r1][i];                  // W
            float m = fminf(a, fminf(g, c));
            // gamma = 1: softmin = m - log(sum exp(m - x)); exponents <= 0
            float s  = __expf(m - a) + __expf(m - g) + __expf(m - c);
            nv = drow[j - 1] + (m - __logf(s));
        }
        R[rn][i] = nv;                           // writes disjoint from reads
        if (tid == 0) R[rn][0] = BIGV;
        __syncthreads();                         // write(d) -> read(d+1)
        int tmp = r2; r2 = r1; r1 = rn; rn = tmp;
        if (d == NPTS + MPTS) result = nv;       // only i==N is valid here
    }
    if (i == NPTS) out[b] = result;
}

// ---------------------------------------------------------------------------
extern "C" void kernel_launch(void* const* d_in, const int* in_sizes, int n_in,
                              void* d_out, int out_size, void* d_ws, size_t ws_size,
                              hipStream_t stream) {
    const float* X = (const float*)d_in[0];      // [32,1024,32] f32
    const float* Y = (const float*)d_in[1];      // [32,1024,32] f32
    float* out = (float*)d_out;                  // [32] f32

    float* x2 = (float*)d_ws;                                   // 32*1024
    float* y2 = x2 + (size_t)BATCH * NPTS;                      // 32*1024
    float* Dm = y2 + (size_t)BATCH * MPTS;                      // 32*1024*1024

    // 1) row norms: 2*B*N threads
    rownorm_kernel<<<(2 * BATCH * NPTS) / 256, 256, 0, stream>>>(X, Y, x2, y2);

    // 2) distance matrix: one wave per 16x16 tile, 8 waves/block
    const int totalTiles = BATCH * (NPTS / 16) * (MPTS / 16);   // 131072
    dist_wmma_kernel<<<totalTiles / 8, 256, 0, stream>>>(X, Y, x2, y2, Dm);

    // 3) soft-DTW DP: one workgroup per batch
    sdtw_kernel<<<BATCH, 1024, 0, stream>>>(Dm, out);
}